// GAT_82265803587630
// MI455X (gfx1250) — compile-verified
//
#include <hip/hip_runtime.h>
#include <hip/hip_bf16.h>
#include <math.h>

typedef float v2f __attribute__((ext_vector_type(2)));
typedef float v8f __attribute__((ext_vector_type(8)));

#define NN   100000
#define EE   3200000
#define INCH 256
#define H1N  4      // heads, layer 1
#define C1N  4      // channels/head, layer 1
#define F1   16     // H1N*C1N
#define OUTC 8

__device__ __forceinline__ float lrelu(float v) { return v > 0.f ? v : 0.2f * v; }
__device__ __forceinline__ float eluf(float v)  { return v > 0.f ? v : __expf(v) - 1.f; }

// monotonic float<->uint encoding so unsigned atomicMax == float max
__device__ __forceinline__ unsigned enc_f32(float f) {
  unsigned u = __float_as_uint(f);
  return (u & 0x80000000u) ? ~u : (u | 0x80000000u);
}
__device__ __forceinline__ float dec_f32(unsigned e) {
  unsigned u = (e & 0x80000000u) ? (e & 0x7fffffffu) : ~e;
  return __uint_as_float(u);
}

// ---------------------------------------------------------------------------
// Layer 1 GEMM: xl1 = x @ W1l, xr1 = x @ W1r  ([N,256]x[256,16])
// One 16-row tile per wave, f32 WMMA 16x16x4, K-loop of 64, weights in LDS.
// ---------------------------------------------------------------------------
__global__ __launch_bounds__(256)
void gat_gemm1(const float* __restrict__ x, const float* __restrict__ W1l,
               const float* __restrict__ W1r, float* __restrict__ xl1,
               float* __restrict__ xr1) {
  __shared__ float wl[INCH * F1];
  __shared__ float wr[INCH * F1];
  for (int i = threadIdx.x; i < INCH * F1; i += 256) {
    wl[i] = W1l[i];
    wr[i] = W1r[i];
  }
  __syncthreads();

  const int wave  = threadIdx.x >> 5;
  const int lane  = threadIdx.x & 31;
  const int tile  = blockIdx.x * 8 + wave;
  if (tile >= NN / 16) return;

  const int col   = lane & 15;          // A: row M, B/C/D: column N
  const int khalf = lane >> 4;          // K sub-pair select / M+8 select
  const int arow  = (tile << 4) + col;

  const float2* __restrict__ ap = (const float2*)(x + (size_t)arow * INCH);

  v8f cl = {};
  v8f cr = {};
#pragma unroll 4
  for (int k = 0; k < INCH; k += 4) {
    // A 16x4 f32: lanes 0-15 hold K={k,k+1}, lanes 16-31 hold K={k+2,k+3}
    float2 av = ap[(k >> 1) + khalf];
    v2f a = {av.x, av.y};
    const int kb = k + 2 * khalf;
    // B 4x16 f32: VGPR0 = K-row kb, VGPR1 = K-row kb+1, column = lane&15
    v2f bl = {wl[kb * F1 + col], wl[(kb + 1) * F1 + col]};
    v2f br = {wr[kb * F1 + col], wr[(kb + 1) * F1 + col]};
    cl = __builtin_amdgcn_wmma_f32_16x16x4_f32(false, a, false, bl, (short)0, cl, false, false);
    cr = __builtin_amdgcn_wmma_f32_16x16x4_f32(false, a, false, br, (short)0, cr, false, false);
  }

#pragma unroll
  for (int v = 0; v < 8; ++v) {
    const int m = (tile << 4) + v + 8 * khalf;   // C/D: VGPR v holds rows v / v+8
    xl1[(size_t)m * F1 + col] = cl[v];
    xr1[(size_t)m * F1 + col] = cr[v];
  }
}

// ---------------------------------------------------------------------------
// Layer 1 edge pass A: logits -> per-dst running max (ordered-uint atomicMax)
// ---------------------------------------------------------------------------
__global__ __launch_bounds__(256)
void gat_edge_max1(const int* __restrict__ ei, const float* __restrict__ xl1,
                   const float* __restrict__ xr1, const float* __restrict__ a1,
                   unsigned* __restrict__ m1) {
  const int e = blockIdx.x * 256 + threadIdx.x;
  if (e >= EE) return;
  __builtin_prefetch(ei + e + 16384, 0, 0);
  __builtin_prefetch(ei + EE + e + 16384, 0, 0);
  const int s = ei[e];
  const int d = ei[EE + e];
  const float4* __restrict__ xs = (const float4*)(xl1 + (size_t)s * F1);
  const float4* __restrict__ xd = (const float4*)(xr1 + (size_t)d * F1);
#pragma unroll
  for (int h = 0; h < H1N; ++h) {
    float4 l = xs[h];
    float4 r = xd[h];
    float4 av = ((const float4*)a1)[h];
    float lg = av.x * lrelu(l.x + r.x) + av.y * lrelu(l.y + r.y) +
               av.z * lrelu(l.z + r.z) + av.w * lrelu(l.w + r.w);
    atomicMax(&m1[(size_t)d * H1N + h], enc_f32(lg));
  }
}

// ---------------------------------------------------------------------------
// Layer 1 edge pass B: recompute logits (gathers are L2-resident), accumulate
// unnormalized numerator agg1[dst] += ex*xl1[src] and denominator den1[dst].
// ---------------------------------------------------------------------------
__global__ __launch_bounds__(256)
void gat_edge_acc1(const int* __restrict__ ei, const float* __restrict__ xl1,
                   const float* __restrict__ xr1, const float* __restrict__ a1,
                   const unsigned* __restrict__ m1, float* __restrict__ den1,
                   float* __restrict__ agg1) {
  const int e = blockIdx.x * 256 + threadIdx.x;
  if (e >= EE) return;
  __builtin_prefetch(ei + e + 16384, 0, 0);
  __builtin_prefetch(ei + EE + e + 16384, 0, 0);
  const int s = ei[e];
  const int d = ei[EE + e];
  const float4* __restrict__ xs = (const float4*)(xl1 + (size_t)s * F1);
  const float4* __restrict__ xd = (const float4*)(xr1 + (size_t)d * F1);
#pragma unroll
  for (int h = 0; h < H1N; ++h) {
    float4 l = xs[h];
    float4 r = xd[h];
    float4 av = ((const float4*)a1)[h];
    float lg = av.x * lrelu(l.x + r.x) + av.y * lrelu(l.y + r.y) +
               av.z * lrelu(l.z + r.z) + av.w * lrelu(l.w + r.w);
    const float ex = __expf(lg - dec_f32(m1[(size_t)d * H1N + h]));
    atomicAdd(&den1[(size_t)d * H1N + h], ex);
    float* ag = agg1 + (size_t)d * F1 + h * C1N;
    atomicAdd(ag + 0, ex * l.x);
    atomicAdd(ag + 1, ex * l.y);
    atomicAdd(ag + 2, ex * l.z);
    atomicAdd(ag + 3, ex * l.w);
  }
}

// ---------------------------------------------------------------------------
// Layer 2 GEMM: h1 = elu(agg1/den1 + b1); xl2 = h1 @ W2l, xr2 = h1 @ W2r
// ([N,16]x[16,8], N-dim zero-padded to 16). normalize+elu fused into A-load.
// ---------------------------------------------------------------------------
__global__ __launch_bounds__(256)
void gat_gemm2(const float* __restrict__ agg1, const float* __restrict__ den1,
               const float* __restrict__ b1, const float* __restrict__ W2l,
               const float* __restrict__ W2r, float* __restrict__ xl2,
               float* __restrict__ xr2) {
  __shared__ float wl[16 * 16];
  __shared__ float wr[16 * 16];
  {
    const int r = threadIdx.x >> 4;
    const int c = threadIdx.x & 15;
    wl[threadIdx.x] = (c < OUTC) ? W2l[r * OUTC + c] : 0.f;
    wr[threadIdx.x] = (c < OUTC) ? W2r[r * OUTC + c] : 0.f;
  }
  __syncthreads();

  const int wave  = threadIdx.x >> 5;
  const int lane  = threadIdx.x & 31;
  const int tile  = blockIdx.x * 8 + wave;
  if (tile >= NN / 16) return;

  const int col   = lane & 15;
  const int khalf = lane >> 4;
  const int arow  = (tile << 4) + col;
  const float* __restrict__ hrow = agg1 + (size_t)arow * F1;
  const float* __restrict__ dn   = den1 + (size_t)arow * H1N;

  v8f cl = {};
  v8f cr = {};
#pragma unroll
  for (int k = 0; k < F1; k += 4) {
    const int kb = k + 2 * khalf;
    const float h0 = eluf(hrow[kb]     / (dn[kb >> 2]       + 1e-16f) + b1[kb]);
    const float h1 = eluf(hrow[kb + 1] / (dn[(kb + 1) >> 2] + 1e-16f) + b1[kb + 1]);
    v2f a = {h0, h1};
    v2f bl = {wl[kb * 16 + col], wl[(kb + 1) * 16 + col]};
    v2f br = {wr[kb * 16 + col], wr[(kb + 1) * 16 + col]};
    cl = __builtin_amdgcn_wmma_f32_16x16x4_f32(false, a, false, bl, (short)0, cl, false, false);
    cr = __builtin_amdgcn_wmma_f32_16x16x4_f32(false, a, false, br, (short)0, cr, false, false);
  }

  if (col < OUTC) {
#pragma unroll
    for (int v = 0; v < 8; ++v) {
      const int m = (tile << 4) + v + 8 * khalf;
      xl2[(size_t)m * OUTC + col] = cl[v];
      xr2[(size_t)m * OUTC + col] = cr[v];
    }
  }
}

// ---------------------------------------------------------------------------
// Layer 2 edge passes (heads=1, 8 channels)
// ---------------------------------------------------------------------------
__device__ __forceinline__ float logit2_of(const float* __restrict__ xl2,
                                           const float* __restrict__ xr2,
                                           const float* __restrict__ a2,
                                           int s, int d) {
  float lg = 0.f;
#pragma unroll
  for (int c = 0; c < OUTC; ++c) {
    lg += a2[c] * lrelu(xl2[(size_t)s * OUTC + c] + xr2[(size_t)d * OUTC + c]);
  }
  return lg;
}

__global__ __launch_bounds__(256)
void gat_edge_max2(const int* __restrict__ ei, const float* __restrict__ xl2,
                   const float* __restrict__ xr2, const float* __restrict__ a2,
                   unsigned* __restrict__ m2) {
  const int e = blockIdx.x * 256 + threadIdx.x;
  if (e >= EE) return;
  __builtin_prefetch(ei + e + 16384, 0, 0);
  __builtin_prefetch(ei + EE + e + 16384, 0, 0);
  const int s = ei[e];
  const int d = ei[EE + e];
  atomicMax(&m2[d], enc_f32(logit2_of(xl2, xr2, a2, s, d)));
}

__global__ __launch_bounds__(256)
void gat_edge_acc2(const int* __restrict__ ei, const float* __restrict__ xl2,
                   const float* __restrict__ xr2, const float* __restrict__ a2,
                   const unsigned* __restrict__ m2, float* __restrict__ den2,
                   float* __restrict__ agg2) {
  const int e = blockIdx.x * 256 + threadIdx.x;
  if (e >= EE) return;
  __builtin_prefetch(ei + e + 16384, 0, 0);
  __builtin_prefetch(ei + EE + e + 16384, 0, 0);
  const int s = ei[e];
  const int d = ei[EE + e];
  const float ex = __expf(logit2_of(xl2, xr2, a2, s, d) - dec_f32(m2[d]));
  atomicAdd(&den2[d], ex);
#pragma unroll
  for (int c = 0; c < OUTC; ++c) {
    atomicAdd(&agg2[(size_t)d * OUTC + c], ex * xl2[(size_t)s * OUTC + c]);
  }
}

// out = sigmoid(agg2/den2 + b2)  (normalization fused here)
__global__ __launch_bounds__(256)
void gat_finalize(const float* __restrict__ agg2, const float* __restrict__ den2,
                  const float* __restrict__ b2, float* __restrict__ out) {
  const int i = blockIdx.x * 256 + threadIdx.x;
  if (i >= NN * OUTC) return;
  const float v = agg2[i] / (den2[i >> 3] + 1e-16f) + b2[i & (OUTC - 1)];
  out[i] = 1.f / (1.f + __expf(-v));
}

// ---------------------------------------------------------------------------
extern "C" void kernel_launch(void* const* d_in, const int* in_sizes, int n_in,
                              void* d_out, int out_size, void* d_ws, size_t ws_size,
                              hipStream_t stream) {
  const float* x   = (const float*)d_in[0];
  const int*   ei  = (const int*)d_in[1];
  const float* W1l = (const float*)d_in[2];
  const float* W1r = (const float*)d_in[3];
  const float* a1  = (const float*)d_in[4];
  const float* b1  = (const float*)d_in[5];
  const float* W2l = (const float*)d_in[6];
  const float* W2r = (const float*)d_in[7];
  const float* a2  = (const float*)d_in[8];
  const float* b2  = (const float*)d_in[9];
  float* out = (float*)d_out;

  // workspace layout (floats) -- ~8.2M floats = 33 MB, no aliasing
  float* wsf = (float*)d_ws;
  const size_t oXL1  = 0;
  const size_t oXR1  = oXL1 + (size_t)NN * F1;
  const size_t oM1   = oXR1 + (size_t)NN * F1;
  const size_t oDEN1 = oM1 + (size_t)NN * H1N;
  const size_t oAGG1 = oDEN1 + (size_t)NN * H1N;
  const size_t oXL2  = oAGG1 + (size_t)NN * F1;
  const size_t oXR2  = oXL2 + (size_t)NN * OUTC;
  const size_t oM2   = oXR2 + (size_t)NN * OUTC;
  const size_t oDEN2 = oM2 + (size_t)NN;
  const size_t oAGG2 = oDEN2 + (size_t)NN;
  const size_t total = oAGG2 + (size_t)NN * OUTC;
  if (ws_size < total * sizeof(float)) return;

  const dim3 blk(256);
  const dim3 gTile((NN / 16 + 7) / 8);   // 782 blocks, 8 tiles (waves) each
  const dim3 gEdge((EE + 255) / 256);    // 12500 blocks
  const dim3 gOut((NN * OUTC + 255) / 256);

  // accumulator init (0 is the minimum of the ordered-uint max encoding)
  hipMemsetAsync(wsf + oM1, 0, (oAGG1 - oM1 + (size_t)NN * F1) * sizeof(float), stream);
  hipMemsetAsync(wsf + oM2, 0, (total - oM2) * sizeof(float), stream);

  // ---- layer 1 ----
  gat_gemm1<<<gTile, blk, 0, stream>>>(x, W1l, W1r, wsf + oXL1, wsf + oXR1);
  gat_edge_max1<<<gEdge, blk, 0, stream>>>(ei, wsf + oXL1, wsf + oXR1, a1,
                                           (unsigned*)(wsf + oM1));
  gat_edge_acc1<<<gEdge, blk, 0, stream>>>(ei, wsf + oXL1, wsf + oXR1, a1,
                                           (const unsigned*)(wsf + oM1),
                                           wsf + oDEN1, wsf + oAGG1);

  // ---- layer 2 ----
  gat_gemm2<<<gTile, blk, 0, stream>>>(wsf + oAGG1, wsf + oDEN1, b1, W2l, W2r,
                                       wsf + oXL2, wsf + oXR2);
  gat_edge_max2<<<gEdge, blk, 0, stream>>>(ei, wsf + oXL2, wsf + oXR2, a2,
                                           (unsigned*)(wsf + oM2));
  gat_edge_acc2<<<gEdge, blk, 0, stream>>>(ei, wsf + oXL2, wsf + oXR2, a2,
                                           (const unsigned*)(wsf + oM2),
                                           wsf + oDEN2, wsf + oAGG2);
  gat_finalize<<<gOut, blk, 0, stream>>>(wsf + oAGG2, wsf + oDEN2, b2, out);
}